// QuantumRNNCell_66185446031729
// MI455X (gfx1250) — compile-verified
//
#include <hip/hip_runtime.h>
#include <math.h>

// ---------------------------------------------------------------------------
// QuantumRNNCell for gfx1250 (MI455X).
//
// Analytic reduction: RZ phases are unit-modulus and cancel in |amp|^2; the
// CNOT ring is GF(2)-linear, so the full 8-qubit sim collapses to
//   zs[b][q] = prod_{q' in MASK[q]} cos(angles[b][q'])
// with MASK = rows of f^3 (f = inverse per-layer CNOT-ring pullback).
// qparams is provably unused.
//
// Pipeline per wave (16 batch rows):
//   GEMM1 (8x v_wmma_f32_16x16x4_f32): angles(16x16, N-padded) = [x|h] @ Wcat^T
//   cos + bias -> LDS transpose -> masked products -> zs(16x8)
//   GEMM2 (2x v_wmma_f32_16x16x4_f32): pre(16x16) = zs @ W_out^T
//   out = tanh(pre + b_out)
// Memory-bound: 12 MB traffic -> ~0.5 us floor at 23.3 TB/s.
// ---------------------------------------------------------------------------

typedef float v2f __attribute__((ext_vector_type(2)));
typedef float v8f __attribute__((ext_vector_type(8)));

__device__ __forceinline__ v2f ld2(const float* p) { return *(const v2f*)p; }

__global__ __launch_bounds__(256) void qrnn_fused_kernel(
    const float* __restrict__ x,     // [B,16]
    const float* __restrict__ h,     // [B,16]
    const float* __restrict__ W_in,  // [8,16]
    const float* __restrict__ b_in,  // [8]
    const float* __restrict__ W_h,   // [8,16]
    const float* __restrict__ b_h,   // [8]
    const float* __restrict__ W_out, // [16,8]
    const float* __restrict__ b_out, // [16]
    float* __restrict__ out)         // [B,16]
{
    // 17-column pad kills LDS bank conflicts on the transposed read.
    __shared__ float ctile[8][16][17];

    const int lane      = threadIdx.x & 31;
    const int waveInBlk = threadIdx.x >> 5;
    const int waveId    = blockIdx.x * (blockDim.x >> 5) + waveInBlk;
    const int rowBase   = waveId * 16;      // 16 batch rows per wave
    const int n         = lane & 15;        // WMMA N-column & tile-row id
    const int hi        = lane >> 4;        // half-wave select (K sub-pair)
    const int nc        = n & 7;            // clamped qubit row (branch-free pad)
    const bool valid_n  = (n < 8);

    // 32-bit element offsets -> GVS addressing (SGPR64 base + VGPR32 offset)
    const int rowOff = (rowBase + n) * 16;  // this lane's A-row start (x/h)

    // ---------- GEMM1: angles = [x|h](16x32) @ [W_in^T ; W_h^T], N padded to 16
    v8f acc = {0.f, 0.f, 0.f, 0.f, 0.f, 0.f, 0.f, 0.f};
#pragma unroll
    for (int kk = 0; kk < 8; ++kk) {
        const int k0 = 4 * kk + 2 * hi;     // lane's K pair in this chunk
        // kk<4 -> x half (k in [0,16)), kk>=4 -> h half; uniform per kk.
        v2f a = (kk < 4) ? ld2(x + rowOff + k0)
                         : ld2(h + rowOff + (k0 - 16));
        // B[k][n] = Wcat[n][k]; load from clamped row, zero-select pad lanes.
        v2f bw = (kk < 4) ? ld2(W_in + nc * 16 + k0)
                          : ld2(W_h  + nc * 16 + (k0 - 16));
        bw = valid_n ? bw : (v2f){0.f, 0.f};   // v_cndmask, no exec branching
        acc = __builtin_amdgcn_wmma_f32_16x16x4_f32(
            /*neg_a=*/false, a, /*neg_b=*/false, bw,
            /*c_mod=*/(short)0, acc, /*reuse_a=*/false, /*reuse_b=*/false);
    }

    // ---------- bias + cos, scatter to LDS: ctile[m][q] = cos(angle[m][q])
    // Unconditional loads (nc always in-range) + value-select: no exec branch.
    const float bias_raw = b_in[nc] + b_h[nc];
    const float bias     = valid_n ? bias_raw : 0.f;
#pragma unroll
    for (int r = 0; r < 8; ++r) {
        const int m = r + 8 * hi;           // batch row within tile
        ctile[waveInBlk][m][n] = __cosf(acc[r] + bias);
    }
    __syncthreads();

    // ---------- masked cosine products: zs[row][q] = prod_{q' in MASK[q]} c[row][q']
    float cr[8];
#pragma unroll
    for (int q = 0; q < 8; ++q) cr[q] = ctile[waveInBlk][n][q];  // lane's row = n

    constexpr unsigned MASK[8] = {0x32u, 0x56u, 0xACu, 0x59u, 0xB3u, 0x66u, 0xCCu, 0x99u};
    float zs[8];
#pragma unroll
    for (int q = 0; q < 8; ++q) {
        float p = 1.f;
#pragma unroll
        for (int qp = 0; qp < 8; ++qp)
            if ((MASK[q] >> qp) & 1u) p *= cr[qp];
        zs[q] = p;
    }

    // ---------- GEMM2: pre = zs(16x8) @ W_out^T(8x16), K=8 -> 2 WMMA chunks
    v8f acc2 = {0.f, 0.f, 0.f, 0.f, 0.f, 0.f, 0.f, 0.f};
#pragma unroll
    for (int kk = 0; kk < 2; ++kk) {
        const int k0 = 4 * kk + 2 * hi;
        v2f a = (v2f){zs[k0], zs[k0 + 1]};     // lane's row = n (dup across halves)
        v2f bw = ld2(W_out + n * 8 + k0);      // B[k][n] = W_out[n][k]
        acc2 = __builtin_amdgcn_wmma_f32_16x16x4_f32(
            false, a, false, bw, (short)0, acc2, false, false);
    }

    // ---------- epilogue: tanh + bias, store. D layout: VGPR r -> (M=r+8*hi, N=n)
    const float bo = b_out[n];
    const int outOff = (rowBase + 8 * hi) * 16 + n;
#pragma unroll
    for (int r = 0; r < 8; ++r) {
        out[outOff + r * 16] = tanhf(acc2[r] + bo);
    }
}

extern "C" void kernel_launch(void* const* d_in, const int* in_sizes, int n_in,
                              void* d_out, int out_size, void* d_ws, size_t ws_size,
                              hipStream_t stream) {
    const float* x     = (const float*)d_in[0];
    const float* h     = (const float*)d_in[1];
    const float* W_in  = (const float*)d_in[2];
    const float* b_in  = (const float*)d_in[3];
    const float* W_h   = (const float*)d_in[4];
    const float* b_h   = (const float*)d_in[5];
    // d_in[6] = qparams: analytically eliminated (unit-modulus phases cancel in |amp|^2)
    const float* W_out = (const float*)d_in[7];
    const float* b_out = (const float*)d_in[8];
    float* out = (float*)d_out;

    const int Btot = in_sizes[0] / 16;          // 65536
    const int rowsPerBlock = 128;               // 8 waves * 16 rows
    const int blocks = (Btot + rowsPerBlock - 1) / rowsPerBlock;  // 512

    qrnn_fused_kernel<<<blocks, 256, 0, stream>>>(
        x, h, W_in, b_in, W_h, b_h, W_out, b_out, out);
}